// distregression_46231027974362
// MI455X (gfx1250) — compile-verified
//
#include <hip/hip_runtime.h>

// distregression: per-pixel triangular hypothesis search.
//   s1, blur: [4, 16, 256, 256] f32.  out: [4, 256, 256] f32.
// Per pixel: S_i = SumLo + 2*PrefB_i, Q_i = SumSqLo + 4*PrefSB_i,
//   argmin_i (Q_i - S_i^2/16)  ->  out = S_{i*}/16.
//
// WMMA mapping (V_WMMA_F32_16X16X4_F32, transposed roles so the argmin is
// lane-local):  A[m][j] = (j < m) ? 1 : 0  (rows = hypotheses, constant),
//               B[j][p] = data             (cols = pixels),
//   D[m][p] = Pref_m[p].  C/D layout puts hypothesis m in the VGPR index and
// pixel p in the lane, so each lane evaluates 8 hypotheses of its own pixel
// entirely in registers; one shfl_xor(16) merges the two hypothesis halves.

typedef float v2f __attribute__((ext_vector_type(2)));
typedef float v8f __attribute__((ext_vector_type(8)));

#define NCH 16
#define HW (256 * 256)

__global__ __launch_bounds__(256) void distreg_wmma_kernel(
    const float* __restrict__ s1, const float* __restrict__ blur,
    float* __restrict__ out) {
  const int lane = threadIdx.x & 31;
  const int wave = threadIdx.x >> 5;
  const int group = blockIdx.x * 8 + wave;   // one 16-pixel group per wave
  const int gpb = HW / 16;                   // groups per batch image
  const int b = group / gpb;
  const int p0 = (group - b * gpb) * 16;     // pixel base within batch
  const int p = lane & 15;                   // this lane's pixel (B/D column)
  const int half = lane >> 4;                // 0: supplies K0/K1, 1: K2/K3
  const int base = b * (NCH * HW) + p0 + p;  // channel-0 element of pixel

  // ---- coalesced loads: this lane covers channels {4c+2*half, +1} ----
  float s1v[8], blv[8], sbv[8];
#pragma unroll
  for (int c = 0; c < 4; ++c) {
    const int j0 = 4 * c + 2 * half;
    s1v[2 * c]     = s1[base + j0 * HW];
    s1v[2 * c + 1] = s1[base + (j0 + 1) * HW];
    blv[2 * c]     = blur[base + j0 * HW];
    blv[2 * c + 1] = blur[base + (j0 + 1) * HW];
  }

  // ---- SumLo, SumSqLo: lane pair (L, L^16) covers the 16 channels ----
  float sl = 0.f, sql = 0.f;
#pragma unroll
  for (int t = 0; t < 8; ++t) {
    sbv[t] = s1v[t] * blv[t];
    const float lo = s1v[t] - blv[t];
    sl += lo;
    sql = fmaf(lo, lo, sql);
  }
  sl += __shfl_xor(sl, 16, 32);   // lane now holds full sums of its pixel p
  sql += __shfl_xor(sql, 16, 32);

  // ---- D[m][p] = Pref_m[p] via 8 WMMAs (4 K-chunks x {blur, s1*blur}) ----
  const int m = lane & 15;  // A-matrix row (hypothesis) this lane supplies
  v8f dB = {0.f, 0.f, 0.f, 0.f, 0.f, 0.f, 0.f, 0.f};
  v8f dSB = {0.f, 0.f, 0.f, 0.f, 0.f, 0.f, 0.f, 0.f};
#pragma unroll
  for (int c = 0; c < 4; ++c) {
    const int j0 = 4 * c + 2 * half;  // global channel rows this lane supplies
    v2f atri = {(j0 < m) ? 1.f : 0.f, (j0 + 1 < m) ? 1.f : 0.f};
    v2f b_b = {blv[2 * c], blv[2 * c + 1]};
    v2f b_sb = {sbv[2 * c], sbv[2 * c + 1]};
    dB = __builtin_amdgcn_wmma_f32_16x16x4_f32(false, atri, false, b_b,
                                               (short)0, dB, false, false);
    dSB = __builtin_amdgcn_wmma_f32_16x16x4_f32(false, atri, false, b_sb,
                                                (short)0, dSB, false, false);
  }

  // ---- lane-local argmin over hypotheses i = r + 8*half (ascending ->
  //      strict '<' reproduces argmin's first-occurrence tie-break) ----
  const float inv16 = 1.0f / 16.0f;
  float bestVar = __builtin_inff();
  float bestS = 0.f;
#pragma unroll
  for (int r = 0; r < 8; ++r) {
    const float S = fmaf(2.f, dB[r], sl);
    const float Q = fmaf(4.f, dSB[r], sql);
    const float var = fmaf(-S * inv16, S, Q);  // ∝ unbiased variance
    if (var < bestVar) { bestVar = var; bestS = S; }
  }
  if (half) {  // hypothesis 16 = Pref_15 + channel-15 term (owned by this lane)
    const float S = fmaf(2.f, dB[7] + blv[7], sl);
    const float Q = fmaf(4.f, dSB[7] + sbv[7], sql);
    const float var = fmaf(-S * inv16, S, Q);
    if (var < bestVar) { bestVar = var; bestS = S; }
  }

  // ---- merge hypothesis halves; low half (i<=7) wins ties ----
  const float oVar = __shfl_xor(bestVar, 16, 32);
  const float oS = __shfl_xor(bestS, 16, 32);
  if (oVar < bestVar || (oVar == bestVar && half)) bestS = oS;

  // ---- one coalesced half-wave store: lane p writes pixel p ----
  if (!half) out[b * HW + p0 + p] = bestS * inv16;
}

extern "C" void kernel_launch(void* const* d_in, const int* in_sizes, int n_in,
                              void* d_out, int out_size, void* d_ws,
                              size_t ws_size, hipStream_t stream) {
  const float* s1 = (const float*)d_in[0];
  const float* blur = (const float*)d_in[1];
  float* out = (float*)d_out;
  // 4*65536/16 = 16384 wave-groups, 8 waves (256 threads) per block
  dim3 grid(2048), block(256);
  hipLaunchKernelGGL(distreg_wmma_kernel, grid, block, 0, stream, s1, blur, out);
}